// OPTAttention_23536420782108
// MI455X (gfx1250) — compile-verified
//
#include <hip/hip_runtime.h>
#include <hip/hip_bf16.h>

// ---------------- problem constants (match reference setup_inputs) ----------
constexpr int B  = 2;
constexpr int H  = 32;
constexpr int Lq = 2048;
constexpr int Lk = 2048;
constexpr int WORDS = Lk / 32;                 // 64 u32 words per 2048-bit mask
constexpr long long NOUT = (long long)B * H * Lq * Lk;

// raw clang vector types (HIP_vector_type structs are rejected by builtins)
typedef int   v4i __attribute__((ext_vector_type(4)));
typedef float v4f __attribute__((ext_vector_type(4)));

// pointer types with explicit address spaces for the async-LDS builtin
typedef __attribute__((address_space(1))) v4i gv4i_t;   // global
typedef __attribute__((address_space(3))) v4i lv4i_t;   // LDS

// ---------------- CDNA5 async global->LDS helpers ---------------------------
__device__ __forceinline__ void async_load_b128(const float* g, void* l) {
#if defined(__gfx1250__) && __has_builtin(__builtin_amdgcn_global_load_async_to_lds_b128)
  __builtin_amdgcn_global_load_async_to_lds_b128(
      (gv4i_t*)g, (lv4i_t*)l, /*offset=*/0, /*cpol=*/0);
#elif defined(__gfx1250__)
  {
    unsigned lds_addr = (unsigned)(unsigned long long)(
        (__attribute__((address_space(3))) void*)l);
    unsigned long long gaddr = (unsigned long long)g;
    asm volatile("global_load_async_to_lds_b128 %0, %1, off"
                 :: "v"(lds_addr), "v"(gaddr) : "memory");
  }
#else
  *(v4f*)l = *(const v4f*)g;   // host-pass / non-gfx1250 fallback
#endif
}

__device__ __forceinline__ void wait_async_lds() {
#if defined(__gfx1250__) && __has_builtin(__builtin_amdgcn_s_wait_asynccnt)
  __builtin_amdgcn_s_wait_asynccnt(0);
#elif defined(__gfx1250__)
  asm volatile("s_wait_asynccnt 0" ::: "memory");
#endif
}

// order-preserving float -> u32 key (monotone: a<b  <=>  key(a)<key(b))
__device__ __forceinline__ unsigned f32_key(unsigned u) {
  return (u & 0x80000000u) ? ~u : (u | 0x80000000u);
}

// ---------------- K1: stream-fill output with zeros (NT stores) -------------
__global__ void k_fill_zero(float* __restrict__ out, long long n4) {
  long long i = (long long)blockIdx.x * blockDim.x + threadIdx.x;
  long long stride = (long long)gridDim.x * blockDim.x;
  v4f z = {0.f, 0.f, 0.f, 0.f};
  v4f* __restrict__ o4 = (v4f*)out;
  for (; i < n4; i += stride) {
#if __has_builtin(__builtin_nontemporal_store)
    __builtin_nontemporal_store(z, &o4[i]);
#else
    o4[i] = z;
#endif
  }
}

// Shared row machinery: loads row into LDS keys, finds k-th-largest key via
// 32-step binary search over the monotone u32 key space, ORs selected bits
// into s_umask. All decisions are block-uniform (driven by an LDS counter).
// ---------------------------------------------------------------------------

// ---------------- K2: per-head union until threshold, record n_h ------------
__global__ void k_topk_union_nh(const float* __restrict__ scores,
                                unsigned* __restrict__ ws_nh,
                                unsigned* __restrict__ ws_masks,
                                int k, int threshold) {
  const int head = blockIdx.x;          // 0 .. B*H-1
  const int tid  = threadIdx.x;         // 0 .. 255
  __shared__ __align__(16) unsigned skeys[Lk];
  __shared__ unsigned s_umask[WORDS];
  __shared__ unsigned s_cnt;

  for (int w = tid; w < WORDS; w += blockDim.x) s_umask[w] = 0u;
  const float* base = scores + (size_t)head * Lq * Lk;

  int nh = 0;
  for (int n = 1; n <= Lq; ++n) {
    const float* rowp = base + (size_t)(Lq - n) * Lk;
    __syncthreads();                                   // LDS row reuse safe
    // async DMA row -> LDS: 2 x b128 per lane (256 lanes * 2 * 16B = 8KB)
    for (int i = 0; i < Lk; i += 4 * 256)
      async_load_b128(rowp + i + tid * 4, &skeys[i + tid * 4]);
    wait_async_lds();
    __syncthreads();
    // convert in place to sortable keys
    for (int i = tid; i < Lk; i += 256) skeys[i] = f32_key(skeys[i]);
    __syncthreads();
    // binary search: largest T with count(key >= T) >= k
    unsigned lo = 0u, hi = 0xFFFFFFFFu;
    while (lo < hi) {
      unsigned mid = lo + ((hi - lo) >> 1) + 1u;
      if (tid == 0) s_cnt = 0u;
      __syncthreads();
      unsigned c = 0;
      for (int i = tid; i < Lk; i += 256) c += (skeys[i] >= mid) ? 1u : 0u;
      if (c) atomicAdd(&s_cnt, c);
      __syncthreads();
      unsigned tot = s_cnt;
      __syncthreads();
      if (tot >= (unsigned)k) lo = mid; else hi = mid - 1u;
    }
    // union top-k bits
    for (int i = tid; i < Lk; i += 256)
      if (skeys[i] >= lo) atomicOr(&s_umask[i >> 5], 1u << (i & 31));
    if (tid == 0) s_cnt = 0u;
    __syncthreads();
    if (tid < WORDS) atomicAdd(&s_cnt, (unsigned)__popc(s_umask[tid]));
    __syncthreads();
    unsigned usz = s_cnt;
    __syncthreads();
    if (usz >= (unsigned)threshold) { nh = n; break; }
  }
  if (nh == 0) nh = Lq;                 // never reached threshold: used all rows
  if (tid == 0) ws_nh[head] = (unsigned)nh;
  for (int w = tid; w < WORDS; w += blockDim.x)
    ws_masks[head * WORDS + w] = s_umask[w];
}

// ---------------- K3: n_stop = max(n_h), reset popsum -----------------------
__global__ void k_reduce_nstop(const unsigned* __restrict__ ws_nh,
                               unsigned* __restrict__ ws_nstop,
                               unsigned* __restrict__ ws_popsum, int nheads) {
  __shared__ unsigned s_max;
  if (threadIdx.x == 0) s_max = 0u;
  __syncthreads();
  if ((int)threadIdx.x < nheads) atomicMax(&s_max, ws_nh[threadIdx.x]);
  __syncthreads();
  if (threadIdx.x == 0) { *ws_nstop = s_max; *ws_popsum = 0u; }
}

// ---------------- K4: extend each head's union to the global n_stop ---------
__global__ void k_extend_union(const float* __restrict__ scores,
                               const unsigned* __restrict__ ws_nh,
                               const unsigned* __restrict__ ws_nstop,
                               unsigned* __restrict__ ws_masks, int k) {
  const int head = blockIdx.x;
  const int tid  = threadIdx.x;
  __shared__ __align__(16) unsigned skeys[Lk];
  __shared__ unsigned s_umask[WORDS];
  __shared__ unsigned s_cnt;

  const int nstop = (int)*ws_nstop;          // uniform scalar load
  const int nh    = (int)ws_nh[head];
  for (int w = tid; w < WORDS; w += blockDim.x)
    s_umask[w] = ws_masks[head * WORDS + w];
  const float* base = scores + (size_t)head * Lq * Lk;

  for (int n = nh + 1; n <= nstop; ++n) {
    const float* rowp = base + (size_t)(Lq - n) * Lk;
    __syncthreads();
    for (int i = 0; i < Lk; i += 4 * 256)
      async_load_b128(rowp + i + tid * 4, &skeys[i + tid * 4]);
    wait_async_lds();
    __syncthreads();
    for (int i = tid; i < Lk; i += 256) skeys[i] = f32_key(skeys[i]);
    __syncthreads();
    unsigned lo = 0u, hi = 0xFFFFFFFFu;
    while (lo < hi) {
      unsigned mid = lo + ((hi - lo) >> 1) + 1u;
      if (tid == 0) s_cnt = 0u;
      __syncthreads();
      unsigned c = 0;
      for (int i = tid; i < Lk; i += 256) c += (skeys[i] >= mid) ? 1u : 0u;
      if (c) atomicAdd(&s_cnt, c);
      __syncthreads();
      unsigned tot = s_cnt;
      __syncthreads();
      if (tot >= (unsigned)k) lo = mid; else hi = mid - 1u;
    }
    for (int i = tid; i < Lk; i += 256)
      if (skeys[i] >= lo) atomicOr(&s_umask[i >> 5], 1u << (i & 31));
    __syncthreads();
  }
  __syncthreads();
  for (int w = tid; w < WORDS; w += blockDim.x)
    ws_masks[head * WORDS + w] = s_umask[w];
}

// ---------------- K5: group-OR, write last query row, accumulate popcount ---
__global__ void k_write_last_rows(const unsigned* __restrict__ ws_masks,
                                  const int* __restrict__ gs_ptr,
                                  float* __restrict__ out,
                                  unsigned* __restrict__ ws_popsum,
                                  float min_value) {
  const int head = blockIdx.x;           // 0 .. B*H-1
  const int tid  = threadIdx.x;
  __shared__ unsigned gm[WORDS];
  __shared__ unsigned s_cnt;
  int gs = *gs_ptr; if (gs < 1) gs = 1; if (gs > H) gs = H;
  const int b  = head / H;
  const int h  = head % H;
  const int g0 = (h / gs) * gs;          // first head of this head's group
  if (tid < WORDS) {
    unsigned m = 0u;
    for (int j = 0; j < gs; ++j) {
      int hh = g0 + j;
      if (hh < H) m |= ws_masks[(b * H + hh) * WORDS + tid];
    }
    gm[tid] = m;
  }
  if (tid == 0) s_cnt = 0u;
  __syncthreads();
  if (tid < WORDS) atomicAdd(&s_cnt, (unsigned)__popc(gm[tid]));
  __syncthreads();
  if (tid == 0) atomicAdd(ws_popsum, s_cnt);   // per-head broadcast popcount
  float* rowp = out + ((size_t)head * Lq + (Lq - 1)) * (size_t)Lk;
  for (int i = tid; i < Lk; i += blockDim.x) {
    bool sel = (gm[i >> 5] >> (i & 31)) & 1u;
    rowp[i] = sel ? 0.0f : min_value;
  }
}

// ---------------- K6: density scalar ----------------------------------------
__global__ void k_density(const unsigned* __restrict__ ws_popsum,
                          float* __restrict__ out_density) {
  *out_density = (float)((double)*ws_popsum /
                         (double)((long long)B * H * Lk));
}

// ---------------- launcher ---------------------------------------------------
extern "C" void kernel_launch(void* const* d_in, const int* in_sizes, int n_in,
                              void* d_out, int out_size, void* d_ws, size_t ws_size,
                              hipStream_t stream) {
  const float* scores = (const float*)d_in[0];
  const int*   gs_ptr = (const int*)d_in[1];     // device scalar group_size
  float*       out    = (float*)d_out;

  unsigned* ws        = (unsigned*)d_ws;
  unsigned* ws_nh     = ws;                      // [64]
  unsigned* ws_nstop  = ws + 64;                 // [1]
  unsigned* ws_popsum = ws + 65;                 // [1]
  unsigned* ws_masks  = ws + 128;                // [64][WORDS] = 16 KB

  // k / threshold exactly per reference arithmetic
  int k = (int)(0.2 * (double)Lk);               // 409
  if (k < 1) k = 1; if (k > Lk) k = Lk;
  int thr = 2 * k;
  int cap = (int)(0.75 * (double)Lk);            // 1536
  if (thr > cap) thr = cap; if (thr < 1) thr = 1;  // 818

  const float MINV = -3.402823466e38f;           // finfo(float32).min

  k_fill_zero<<<8192, 256, 0, stream>>>(out, NOUT / 4);
  k_topk_union_nh<<<B * H, 256, 0, stream>>>(scores, ws_nh, ws_masks, k, thr);
  k_reduce_nstop<<<1, 64, 0, stream>>>(ws_nh, ws_nstop, ws_popsum, B * H);
  k_extend_union<<<B * H, 256, 0, stream>>>(scores, ws_nh, ws_nstop, ws_masks, k);
  k_write_last_rows<<<B * H, 256, 0, stream>>>(ws_masks, gs_ptr, out, ws_popsum, MINV);
  k_density<<<1, 1, 0, stream>>>(ws_popsum, out + NOUT);
}